// CustomLoss_38577396253443
// MI455X (gfx1250) — compile-verified
//
#include <hip/hip_runtime.h>
#include <hip/hip_bf16.h>
#include <math.h>

#define NBOX 1024

typedef float v2f __attribute__((ext_vector_type(2)));
typedef float v8f __attribute__((ext_vector_type(8)));

// ---------------------------------------------------------------------------
// Kernel 1: build IoU matrix JM[1024][1024]; rows = labels, cols = predictions.
// One wave32 per 16x16 tile. Denominator formed by V_WMMA_F32_16X16X4_F32:
//   D = A(16x4) x B(4x16) + C,  A[m]=[area_a[m],1,0,0], B[:,n]=[1,area_b[n],0,0]
//   C[m][n] = -inter[m][n]   =>  D = area_a + area_b - inter
// IoU = inter / D. The K-sum is invariant to K-lane mapping since A and B use
// the same convention.
// ---------------------------------------------------------------------------
__global__ __launch_bounds__(256) void jm_build_kernel(
    const float* __restrict__ labels, const float* __restrict__ preds,
    float* __restrict__ JM)
{
  const int wavesPerBlock = 256 / 32;
  int tile = blockIdx.x * wavesPerBlock + ((int)threadIdx.x >> 5);
  int ti = tile >> 6;        // row tile (labels),   0..63
  int tj = tile & 63;        // col tile (predictions), 0..63
  int lane = threadIdx.x & 31;
  int half = lane >> 4;      // 0: lanes 0-15, 1: lanes 16-31
  int ln   = lane & 15;

  // This lane's prediction box (column n = ln of the tile)
  int col = tj * 16 + ln;
  float bx1 = preds[col * 4 + 0], by1 = preds[col * 4 + 1];
  float bx2 = preds[col * 4 + 2], by2 = preds[col * 4 + 3];
  float area_b = (bx2 - bx1) * (by2 - by1);

  // A-operand row m = ln (A's M index is lane%16 in both lane halves)
  int am = ti * 16 + ln;
  float ax1 = labels[am * 4 + 0], ay1 = labels[am * 4 + 1];
  float ax2 = labels[am * 4 + 2], ay2 = labels[am * 4 + 3];
  float area_a = (ax2 - ax1) * (ay2 - ay1);

  // A: K=0 -> area_a, K=1 -> 1, K=2,3 -> 0 (upper lane-half holds K=2,3)
  // B: K=0 -> 1, K=1 -> area_b, K=2,3 -> 0
  v2f A, B;
  A.x = half ? 0.0f : area_a;
  A.y = half ? 0.0f : 1.0f;
  B.x = half ? 0.0f : 1.0f;
  B.y = half ? 0.0f : area_b;

  // C/D layout (32-bit 16x16): VGPR v, lanes0-15 -> M=v, lanes16-31 -> M=v+8
  v8f C;
  float inter[8];
#pragma unroll
  for (int v = 0; v < 8; ++v) {
    int row = ti * 16 + v + 8 * half;
    float lx1 = labels[row * 4 + 0], ly1 = labels[row * 4 + 1];
    float lx2 = labels[row * 4 + 2], ly2 = labels[row * 4 + 3];
    float iw = fminf(lx2, bx2) - fmaxf(lx1, bx1);
    float ih = fminf(ly2, by2) - fmaxf(ly1, by1);
    iw = fmaxf(iw, 0.0f);
    ih = fmaxf(ih, 0.0f);
    inter[v] = iw * ih;
    C[v] = -inter[v];
  }

  // D = A*B + C  (emits v_wmma_f32_16x16x4_f32; EXEC is all 1s here)
  v8f D = __builtin_amdgcn_wmma_f32_16x16x4_f32(
      /*neg_a=*/false, A, /*neg_b=*/false, B,
      /*c_mod=*/(short)0, C, /*reuse_a=*/false, /*reuse_b=*/false);

#pragma unroll
  for (int v = 0; v < 8; ++v) {
    int row = ti * 16 + v + 8 * half;
    JM[row * NBOX + col] = inter[v] / D[v];
  }
}

// ---------------------------------------------------------------------------
// Kernel 2: sequential greedy matching (1024 steps) + final mean-L1 loss.
// Single workgroup of 1024 threads (32 wave32s) on one WGP; all matching
// state in LDS. Incremental argmax: per-row cached (max, arg); each step only
// recomputes rows invalidated by the newly masked column / swapped row.
// Tie-breaking reproduces the reference's "last row-major occurrence":
// value first, then larger row index, then larger column index.
// ---------------------------------------------------------------------------
__global__ __launch_bounds__(1024) void greedy_match_kernel(
    const float* __restrict__ preds, const float* __restrict__ labels,
    const float* __restrict__ JM, float* __restrict__ out)
{
  __shared__ float rowmax[NBOX];
  __shared__ int   rowarg[NBOX];
  __shared__ int   perm[NBOX];
  __shared__ int   seen[NBOX];
  __shared__ float redv[32];
  __shared__ int   redi[32];
  __shared__ int   list[NBOX];
  __shared__ int   nrec;
  __shared__ int   selR, selC;

  const int t    = threadIdx.x;
  const int wave = t >> 5;
  const int lane = t & 31;

  perm[t] = t;
  seen[t] = 0;
  __syncthreads();

  // Initial per-row maxima (perm == identity): one wave per row, coalesced.
  for (int i = wave; i < NBOX; i += 32) {
    const float* row = JM + (size_t)i * NBOX;
    float bv = -1.0f;
    int   bj = -1;
    for (int j = lane; j < NBOX; j += 32) {
      __builtin_prefetch(&row[j + 256], 0, 0);  // global_prefetch_b8
      float v = row[j];
      if (v >= bv) { bv = v; bj = j; }          // per-lane j increases: '>=' keeps last
    }
    for (int off = 16; off > 0; off >>= 1) {
      float ov = __shfl_down(bv, off);
      int   oj = __shfl_down(bj, off);
      if (ov > bv || (ov == bv && oj > bj)) { bv = ov; bj = oj; }
    }
    if (lane == 0) { rowmax[i] = bv; rowarg[i] = bj; }
  }
  __syncthreads();

  for (int step = 0; step < NBOX; ++step) {
    // ---- Phase 1: argmax over active rows (value, then larger row index)
    float v = seen[t] ? -3.0f : rowmax[t];
    int   i = t;
    for (int off = 16; off > 0; off >>= 1) {
      float ov = __shfl_down(v, off);
      int   oi = __shfl_down(i, off);
      if (ov > v || (ov == v && oi > i)) { v = ov; i = oi; }
    }
    if (lane == 0) { redv[wave] = v; redi[wave] = i; }
    __syncthreads();
    if (wave == 0) {
      float v2 = redv[lane];
      int   i2 = redi[lane];
      for (int off = 16; off > 0; off >>= 1) {
        float ov = __shfl_down(v2, off);
        int   oi = __shfl_down(i2, off);
        if (ov > v2 || (ov == v2 && oi > i2)) { v2 = ov; i2 = oi; }
      }
      if (lane == 0) {
        int r = i2;
        int c = rowarg[r];
        selR = r;
        selC = c;
        int pr = perm[r], pc = perm[c];
        perm[r] = pc;
        perm[c] = pr;
        seen[c] = 1;
        nrec = 0;
      }
    }
    __syncthreads();

    // ---- Phase 2: collect rows needing recompute (swapped row r, and any
    //      active row whose cached argmax was the newly masked column c)
    int r = selR;
    int c = selC;
    if (!seen[t] && (t == r || rowarg[t] == c)) {
      int slot = atomicAdd(&nrec, 1);
      list[slot] = t;
    }
    __syncthreads();

    // ---- Phase 3: recompute those rows, one wave per row
    int nr = nrec;
    for (int q = wave; q < nr; q += 32) {
      int ri = list[q];
      const float* row = JM + (size_t)perm[ri] * NBOX;
      float bv = -1.0f;
      int   bj = -1;
      for (int j = lane; j < NBOX; j += 32) {
        __builtin_prefetch(&row[j + 256], 0, 0);
        if (!seen[j]) {
          float vv = row[j];
          if (vv >= bv) { bv = vv; bj = j; }
        }
      }
      for (int off = 16; off > 0; off >>= 1) {
        float ov = __shfl_down(bv, off);
        int   oj = __shfl_down(bj, off);
        if (ov > bv || (ov == bv && oj > bj)) { bv = ov; bj = oj; }
      }
      if (lane == 0) { rowmax[ri] = bv; rowarg[ri] = bj; }
    }
    __syncthreads();
  }

  // ---- Epilogue: loss = mean(|labels[perm] - predictions|) over 4096 elems
  float s = 0.0f;
  int p = perm[t];
#pragma unroll
  for (int k = 0; k < 4; ++k)
    s += fabsf(labels[p * 4 + k] - preds[t * 4 + k]);
  for (int off = 16; off > 0; off >>= 1) s += __shfl_down(s, off);
  if (lane == 0) redv[wave] = s;
  __syncthreads();
  if (t == 0) {
    float tot = 0.0f;
    for (int w = 0; w < 32; ++w) tot += redv[w];
    out[0] = tot / (float)(NBOX * 4);
  }
}

// ---------------------------------------------------------------------------
extern "C" void kernel_launch(void* const* d_in, const int* in_sizes, int n_in,
                              void* d_out, int out_size, void* d_ws, size_t ws_size,
                              hipStream_t stream) {
  (void)in_sizes; (void)n_in; (void)out_size; (void)ws_size;
  const float* preds  = (const float*)d_in[0];  // predictions [1024,4]
  const float* labels = (const float*)d_in[1];  // labels      [1024,4]
  // d_in[2] = scores: unused (reference discards the cross-entropy term)

  float* JM = (float*)d_ws;  // 1024*1024*4 = 4 MB scratch (L2-resident)

  // 64x64 tiles of 16x16, 8 waves per 256-thread block -> 512 blocks
  jm_build_kernel<<<512, 256, 0, stream>>>(labels, preds, JM);

  // Single 1024-thread workgroup runs the serial greedy loop + loss
  greedy_match_kernel<<<1, 1024, 0, stream>>>(preds, labels, JM, (float*)d_out);
}